// EcgPreprocess_70411693851001
// MI455X (gfx1250) — compile-verified
//
#include <hip/hip_runtime.h>
#include <hip/hip_bf16.h>

// ECG baseline removal: y = x - med301(med101(x)), exact element medians.
// Per-block chunk of one row staged into LDS via gfx1250 async-to-LDS DMA,
// converted in-place to order-preserving uint32, then per-output exact k-th
// smallest via 2-bit radix selection over the window (16 passes x K ds_loads).

#define T_LEN 5000
#define CHUNK 500            // 5000 / 500 = 10 chunks per row
#define ROWS  96             // 8 * 12

__device__ __forceinline__ unsigned f2ord_bits(unsigned u) {
    // monotone map: float order -> unsigned order (3 VALU ops)
    return u ^ ((unsigned)(((int)u) >> 31) | 0x80000000u);
}

__device__ __forceinline__ float ord2f(unsigned u) {
    return (u & 0x80000000u) ? __uint_as_float(u ^ 0x80000000u)
                             : __uint_as_float(~u);
}

template <int K, int P, bool SUB>
__global__ void median_chunk_kernel(const float* __restrict__ in,
                                    const float* __restrict__ orig,
                                    float* __restrict__ out) {
    // LDS: chunk + halo on both sides
    __shared__ unsigned s[CHUNK + 2 * P];

    const int row    = blockIdx.y;                 // 0..95
    const int c0     = blockIdx.x * CHUNK;         // chunk start (global t)
    const float* rin = in + (size_t)row * T_LEN;

    // --- Stage [c0 - P, c0 + CHUNK + P) with edge-replicate clamping via
    //     gfx1250 async global->LDS DMA (ASYNCcnt-tracked, no VGPR bounce).
    for (int i = threadIdx.x; i < CHUNK + 2 * P; i += blockDim.x) {
        int src = c0 + i - P;
        src = src < 0 ? 0 : (src > T_LEN - 1 ? T_LEN - 1 : src);
        const float* gptr = rin + src;
        unsigned lds_off = (unsigned)(unsigned long long)(&s[i]); // addr[31:0] = LDS byte offset
        asm volatile("global_load_async_to_lds_b32 %0, %1, off"
                     :
                     : "v"(lds_off), "v"(gptr)
                     : "memory");
    }
    // Wait for this wave's async copies, then make all waves' copies visible.
    asm volatile("s_wait_asynccnt 0x0" ::: "memory");
    __syncthreads();

    // --- In-place transform to order-preserving uint encoding.
    for (int i = threadIdx.x; i < CHUNK + 2 * P; i += blockDim.x) {
        s[i] = f2ord_bits(s[i]);
    }
    __syncthreads();

    // --- Each thread computes outputs t = c0 + local, local in [0, CHUNK)
    for (int local = threadIdx.x; local < CHUNK; local += blockDim.x) {
        const unsigned* __restrict__ w = s + local;  // window of K elements
        unsigned pref = 0u;
        int r = K / 2;  // 0-based median rank for odd K

        #pragma unroll 1
        for (int b = 30; b >= 0; b -= 2) {
            // mask of bits strictly above the pair [b+1:b]
            const unsigned maskAbove = (~3u) << b;
            int cnt0 = 0, cnt1 = 0, cnt2 = 0;
            #pragma unroll 4
            for (int j = 0; j < K; ++j) {
                unsigned v = w[j];
                bool m = ((v ^ pref) & maskAbove) == 0u;
                unsigned d = (v >> b) & 3u;
                cnt0 += (int)(m & (d == 0u));
                cnt1 += (int)(m & (d == 1u));
                cnt2 += (int)(m & (d == 2u));
            }
            unsigned two;
            if (r < cnt0) {
                two = 0u;
            } else {
                r -= cnt0;
                if (r < cnt1) {
                    two = 1u;
                } else {
                    r -= cnt1;
                    if (r < cnt2) { two = 2u; }
                    else          { r -= cnt2; two = 3u; }
                }
            }
            pref |= two << b;
        }

        const float med = ord2f(pref);
        const int t = c0 + local;
        if (SUB) {
            out[(size_t)row * T_LEN + t] = orig[(size_t)row * T_LEN + t] - med;
        } else {
            out[(size_t)row * T_LEN + t] = med;
        }
    }
}

extern "C" void kernel_launch(void* const* d_in, const int* in_sizes, int n_in,
                              void* d_out, int out_size, void* d_ws, size_t ws_size,
                              hipStream_t stream) {
    (void)in_sizes; (void)n_in; (void)out_size; (void)ws_size;

    const float* x  = (const float*)d_in[0];     // (8,12,5000) f32
    float* out      = (float*)d_out;             // (8,12,5000) f32
    float* y1       = (float*)d_ws;              // intermediate, 96*5000 f32

    dim3 grid(T_LEN / CHUNK, ROWS);              // (10, 96) = 960 blocks
    dim3 block(256);                             // 8 wave32 per block

    // Pass 1: median window 101 (p=50):  x -> y1
    median_chunk_kernel<101, 50, false><<<grid, block, 0, stream>>>(x, nullptr, y1);
    // Pass 2: median window 301 (p=150): y1 -> median, out = x - median
    median_chunk_kernel<301, 150, true><<<grid, block, 0, stream>>>(y1, x, out);
}